// ISNEModel_43044162241109
// MI455X (gfx1250) — compile-verified
//
#include <hip/hip_runtime.h>
#include <stdint.h>

// ISNE gather + masked-mean kernel for MI455X (gfx1250).
// Memory-bound: ~277 MB random 512B row gathers + 25 MB streaming writes.
// Strategy: one wave32 per batch element; async global->LDS gathers
// (GLOBAL_LOAD_ASYNC_TO_LDS_B128, ASYNCcnt) keep up to ~64 row gathers in
// flight per wave with no VGPR landing pressure. Output stores are
// non-temporal (TH_STORE_NT) so the 25 MB of streaming writes do not evict
// the randomly-gathered 512 MB table from the 192 MB L2.

typedef float v4f __attribute__((ext_vector_type(4)));

constexpr int D          = 128;      // embed dim
constexpr int KMAX       = 32;       // max neighbors (== wave32 width)
constexpr int ROW_BYTES  = D * 4;    // 512 B per row
constexpr int WAVES      = 4;        // waves per block
constexpr int BLOCK      = WAVES * 32;
// per-wave staging: pos(32 rows) + neg(32 rows) = 32 KB; block = 128 KB
constexpr int WAVE_LDS   = 2 * KMAX * ROW_BYTES;

__device__ __forceinline__ void async_row_piece(uint32_t lds_addr, const void* g) {
    // Each lane copies 16 bytes from global address g to the LDS address in lds_addr; tracked by ASYNCcnt.
    asm volatile("global_load_async_to_lds_b128 %0, %1, off"
                 :: "v"(lds_addr), "v"(g)
                 : "memory");
}

__device__ __forceinline__ void wait_async_all() {
    asm volatile("s_wait_asynccnt 0x0" ::: "memory");
}

__global__ __launch_bounds__(BLOCK)
void isne_gather_kernel(const float* __restrict__ table,
                        const int*  __restrict__ node_indices,
                        const int*  __restrict__ pos_idx,
                        const int*  __restrict__ pos_len,
                        const int*  __restrict__ neg_idx,
                        const int*  __restrict__ neg_len,
                        float* __restrict__ out,
                        int batch) {
    __shared__ __align__(16) unsigned char smem[WAVES * WAVE_LDS];

    const int lane = threadIdx.x & 31;
    const int wave = threadIdx.x >> 5;
    const int b    = blockIdx.x * WAVES + wave;
    if (b >= batch) return;

    const uint32_t lds_base = (uint32_t)(uintptr_t)(&smem[0]) + (uint32_t)wave * WAVE_LDS;
    const uint32_t pos_lds  = lds_base;
    const uint32_t neg_lds  = lds_base + KMAX * ROW_BYTES;

    int len_p = pos_len[b]; if (len_p > KMAX) len_p = KMAX; if (len_p < 0) len_p = 0;
    int len_n = neg_len[b]; if (len_n > KMAX) len_n = KMAX; if (len_n < 0) len_n = 0;

    // K == 32 == wave width: one coalesced load grabs the whole index row.
    const int myp   = pos_idx[(size_t)b * KMAX + lane];
    const int myn   = neg_idx[(size_t)b * KMAX + lane];
    const int nodei = node_indices[b];

    // ---- Issue ALL row gathers asynchronously (<= 64 in flight per wave) ----
    for (int k = 0; k < len_p; ++k) {
        const int r = __shfl(myp, k);
        const v4f* g = reinterpret_cast<const v4f*>(table + (size_t)r * D) + lane;
        async_row_piece(pos_lds + (uint32_t)k * ROW_BYTES + (uint32_t)lane * 16, g);
    }
    for (int k = 0; k < len_n; ++k) {
        const int r = __shfl(myn, k);
        const v4f* g = reinterpret_cast<const v4f*>(table + (size_t)r * D) + lane;
        async_row_piece(neg_lds + (uint32_t)k * ROW_BYTES + (uint32_t)lane * 16, g);
    }

    // Node row: plain vector load (RT policy: table lines should stay cached),
    // overlaps the async traffic via LOADcnt.
    const v4f nv = reinterpret_cast<const v4f*>(table + (size_t)nodei * D)[lane];

    // ---- Drain async gathers, then reduce from LDS ----
    wait_async_all();

    const v4f* pbuf =
        reinterpret_cast<const v4f*>(smem + (size_t)wave * WAVE_LDS);
    const v4f* nbuf =
        reinterpret_cast<const v4f*>(smem + (size_t)wave * WAVE_LDS + KMAX * ROW_BYTES);

    v4f pacc = (v4f)0.0f;
    for (int k = 0; k < len_p; ++k) {
        pacc += pbuf[k * 32 + lane];
    }
    v4f nacc = (v4f)0.0f;
    for (int k = 0; k < len_n; ++k) {
        nacc += nbuf[k * 32 + lane];
    }

    const float invp = (len_p > 0) ? (1.0f / (float)len_p) : 0.0f;
    const float invn = (len_n > 0) ? (1.0f / (float)len_n) : 0.0f;
    const v4f pr = pacc * invp;
    const v4f nr = nacc * invn;

    // Outputs concatenated flat: [node | pos | neg], each [B, 128].
    // Non-temporal stores: pure streaming output, keep L2 for the table.
    v4f* o = reinterpret_cast<v4f*>(out);
    constexpr int VPER = D / 4;  // 32 x v4f per row == one per lane
    __builtin_nontemporal_store(nv, &o[(size_t)b * VPER + lane]);
    __builtin_nontemporal_store(pr, &o[((size_t)batch + b) * VPER + lane]);
    __builtin_nontemporal_store(nr, &o[((size_t)(2 * batch) + b) * VPER + lane]);
}

extern "C" void kernel_launch(void* const* d_in, const int* in_sizes, int n_in,
                              void* d_out, int out_size, void* d_ws, size_t ws_size,
                              hipStream_t stream) {
    const float* table        = (const float*)d_in[0];
    const int*   node_indices = (const int*)d_in[1];
    const int*   pos_idx      = (const int*)d_in[2];
    const int*   pos_len      = (const int*)d_in[3];
    const int*   neg_idx      = (const int*)d_in[4];
    const int*   neg_len      = (const int*)d_in[5];
    float*       out          = (float*)d_out;

    const int batch = in_sizes[1];  // node_indices element count

    dim3 block(BLOCK);
    dim3 grid((batch + WAVES - 1) / WAVES);
    isne_gather_kernel<<<grid, block, 0, stream>>>(
        table, node_indices, pos_idx, pos_len, neg_idx, neg_len, out, batch);
}